// CustomLoss_10333691314204
// MI455X (gfx1250) — compile-verified
//
#include <hip/hip_runtime.h>

// ---------------------------------------------------------------------------
// loss = (1/B) * sum_{b,c} (output-target)[b,c] * (|dist[c]-dist[argmax_c target[b,:]]|+1)
// B = 4194304, C = 5.  Pure HBM-bound streaming reduction (~168 MB @ 23.3 TB/s).
// ---------------------------------------------------------------------------

#define NBLOCKS 1024
#define NTHREADS 256
constexpr int kBatch  = 4194304;          // B
constexpr int kGroups = kBatch / 4;       // 4 rows (20 floats) per group

typedef __attribute__((ext_vector_type(4))) float v4f;
typedef __attribute__((ext_vector_type(2))) float v2f;
typedef __attribute__((ext_vector_type(8))) float v8f;

// ---------------------------------------------------------------------------
// Full wave32 sum using V_WMMA_F32_16X16X4_F32 (fp32-exact).
// Step 1:  A[m,0]=v_m (lanes 0-15, VGPR0), A[m,2]=v_{m+16} (lanes 16-31, VGPR0),
//          A[*,1]=A[*,3]=0, B=ones  ->  D1[m,n] = v_m + v_{m+16}  (all n equal).
// Step 2:  D1's 8 C/D VGPRs (rows j & j+8 per VGPR) fed back as four A pairs,
//          B=ones, C-chained      ->  every element = sum of all 16 row-sums
//                                   = full 32-lane sum, in every lane.
// EXEC must be all 1s at the call site (uniform control flow guaranteed below).
// ---------------------------------------------------------------------------
__device__ __forceinline__ float wave_reduce_wmma(float v) {
  v2f ones; ones[0] = 1.0f; ones[1] = 1.0f;
  v2f a;    a[0] = v;       a[1] = 0.0f;
  v8f zero = {};
  v8f d1 = __builtin_amdgcn_wmma_f32_16x16x4_f32(
      false, a, false, ones, (short)0, zero, false, false);

  v8f acc = {};
  v2f a2;
  a2[0] = d1[0]; a2[1] = d1[1];
  acc = __builtin_amdgcn_wmma_f32_16x16x4_f32(
      false, a2, false, ones, (short)0, acc, false, false);
  a2[0] = d1[2]; a2[1] = d1[3];
  acc = __builtin_amdgcn_wmma_f32_16x16x4_f32(
      false, a2, false, ones, (short)0, acc, false, false);
  a2[0] = d1[4]; a2[1] = d1[5];
  acc = __builtin_amdgcn_wmma_f32_16x16x4_f32(
      false, a2, false, ones, (short)0, acc, false, false);
  a2[0] = d1[6]; a2[1] = d1[7];
  acc = __builtin_amdgcn_wmma_f32_16x16x4_f32(
      false, a2, false, ones, (short)0, acc, false, false);
  return acc[0];
}

__global__ void __launch_bounds__(NTHREADS)
loss_partial_kernel(const float* __restrict__ out,
                    const float* __restrict__ tgt,
                    const float* __restrict__ dist,
                    float* __restrict__ partials) {
  // distances are tiny & uniform -> scalar cached loads
  float dd[5];
  #pragma unroll
  for (int c = 0; c < 5; ++c) dd[c] = dist[c];

  const v4f* __restrict__ op = (const v4f*)out;
  const v4f* __restrict__ tp = (const v4f*)tgt;

  const int tid    = blockIdx.x * NTHREADS + threadIdx.x;
  const int stride = NBLOCKS * NTHREADS;        // 262144 -> exactly 4 iters/thread

  float acc = 0.0f;
  for (int g = tid; g < kGroups; g += stride) { // uniform trip count, EXEC stays full
    v4f tv[5], ov[5];
    #pragma unroll
    for (int j = 0; j < 5; ++j) {               // 4 rows = 80 B = 5 x b128, 16B aligned
      tv[j] = __builtin_nontemporal_load(tp + (long)g * 5 + j);
      ov[j] = __builtin_nontemporal_load(op + (long)g * 5 + j);
    }
    float tb[20], ob[20];
    #pragma unroll
    for (int j = 0; j < 5; ++j) {
      #pragma unroll
      for (int k = 0; k < 4; ++k) {
        tb[4 * j + k] = tv[j][k];
        ob[4 * j + k] = ov[j][k];
      }
    }
    #pragma unroll
    for (int r = 0; r < 4; ++r) {
      // argmax with first-index tie rule (strict >)
      float best  = tb[5 * r];
      float delta = dd[0];
      #pragma unroll
      for (int c = 1; c < 5; ++c) {
        const bool gt = tb[5 * r + c] > best;
        best  = gt ? tb[5 * r + c] : best;
        delta = gt ? dd[c] : delta;
      }
      #pragma unroll
      for (int c = 0; c < 5; ++c) {
        const float e = fabsf(dd[c] - delta) + 1.0f;
        acc = fmaf(ob[5 * r + c] - tb[5 * r + c], e, acc);
      }
    }
  }

  // wave32 sum via WMMA, then 8 wave sums via LDS
  const float w = wave_reduce_wmma(acc);
  __shared__ float wsum[NTHREADS / 32];
  if ((threadIdx.x & 31) == 0) wsum[threadIdx.x >> 5] = w;
  __syncthreads();
  if (threadIdx.x == 0) {
    float t = 0.0f;
    #pragma unroll
    for (int i = 0; i < NTHREADS / 32; ++i) t += wsum[i];
    partials[blockIdx.x] = t;
  }
}

__global__ void __launch_bounds__(NTHREADS)
loss_final_kernel(const float* __restrict__ partials, float* __restrict__ result) {
  float acc = 0.0f;
  for (int i = threadIdx.x; i < NBLOCKS; i += NTHREADS)  // uniform: 4 iters/thread
    acc += partials[i];
  const float w = wave_reduce_wmma(acc);
  __shared__ float wsum[NTHREADS / 32];
  if ((threadIdx.x & 31) == 0) wsum[threadIdx.x >> 5] = w;
  __syncthreads();
  if (threadIdx.x == 0) {
    float t = 0.0f;
    #pragma unroll
    for (int i = 0; i < NTHREADS / 32; ++i) t += wsum[i];
    result[0] = t * (1.0f / (float)kBatch);
  }
}

extern "C" void kernel_launch(void* const* d_in, const int* in_sizes, int n_in,
                              void* d_out, int out_size, void* d_ws, size_t ws_size,
                              hipStream_t stream) {
  (void)in_sizes; (void)n_in; (void)out_size; (void)ws_size;
  const float* output   = (const float*)d_in[0];   // [B,5] fp32
  const float* target   = (const float*)d_in[1];   // [B,5] fp32
  const float* distance = (const float*)d_in[2];   // [5]   fp32
  float* partials = (float*)d_ws;                  // NBLOCKS floats of scratch
  float* result   = (float*)d_out;                 // 1 float

  loss_partial_kernel<<<NBLOCKS, NTHREADS, 0, stream>>>(output, target, distance, partials);
  loss_final_kernel<<<1, NTHREADS, 0, stream>>>(partials, result);
}